// TTlinear_31001073942593
// MI455X (gfx1250) — compile-verified
//
#include <hip/hip_runtime.h>

// TT-linear on MI455X (gfx1250):
//   Step 1: collapse the 4 TT cores into dense W[4096][4096] (~0.35 GFLOP, 3 tiny kernels)
//   Step 2: Y = X @ W^T + bias as one 4096^3 fp32 GEMM on V_WMMA_F32_16X16X4_F32
// X + W + Y = 192 MB ~= global L2, so the GEMM is matrix-pipe bound, not HBM bound.

typedef float v2f __attribute__((ext_vector_type(2)));
typedef float v4f __attribute__((ext_vector_type(4)));
typedef float v8f __attribute__((ext_vector_type(8)));

#define DIN   4096
#define LDSTR 36   // padded LDS row stride (floats): keeps 16B alignment, spreads banks

// ---------------------------------------------------------------------------
// Step 1a: G01[p=o0*8+i0][q=o1*8+i1][r2] = sum_r1 f0[p][r1] * f1[r1][o1][i1][r2]
// 65536 elements, one thread each.
__global__ __launch_bounds__(256) void build_g01(const float* __restrict__ f0,
                                                 const float* __restrict__ f1,
                                                 float* __restrict__ g01) {
    int t  = blockIdx.x * 256 + threadIdx.x;      // [0, 65536)
    int r2 = t & 15;
    int q  = (t >> 4) & 63;
    int p  = t >> 10;
    int o1 = q >> 3, i1 = q & 7;
    float s = 0.f;
#pragma unroll
    for (int r1 = 0; r1 < 16; ++r1)
        s += f0[p * 16 + r1] * f1[((r1 * 8 + o1) * 8 + i1) * 16 + r2];
    g01[t] = s;   // flat index == t by construction
}

// Step 1b: G012[a=(p*64+q)][b=o2*8+i2][r3] = sum_r2 G01[a][r2] * f2[r2][o2][i2][r3]
// 4096*64*16 = 4.19M elements.
__global__ __launch_bounds__(256) void build_g012(const float* __restrict__ g01,
                                                  const float* __restrict__ f2,
                                                  float* __restrict__ g012) {
    int t  = blockIdx.x * 256 + threadIdx.x;      // [0, 4194304)
    int r3 = t & 15;
    int b  = (t >> 4) & 63;
    int a  = t >> 10;
    float s = 0.f;
#pragma unroll
    for (int r2 = 0; r2 < 16; ++r2)
        s += g01[a * 16 + r2] * f2[(r2 * 64 + b) * 16 + r3];
    g012[t] = s;
}

// Step 1c: W[o][i] = sum_r3 G012[a][b][r3] * f3[r3][o3][i3]
//   o = o0*512+o1*64+o2*8+o3,  i = i0*512+i1*64+i2*8+i3
__global__ __launch_bounds__(256) void build_w(const float* __restrict__ g012,
                                               const float* __restrict__ f3,
                                               float* __restrict__ W) {
    int t = blockIdx.x * 256 + threadIdx.x;       // [0, 16777216)
    int o = t >> 12, i = t & 4095;
    int o0 = o >> 9, o1 = (o >> 6) & 7, o2 = (o >> 3) & 7, o3 = o & 7;
    int i0 = i >> 9, i1 = (i >> 6) & 7, i2 = (i >> 3) & 7, i3 = i & 7;
    int a = ((o0 * 8 + i0) * 8 + o1) * 8 + i1;    // [0,4096)
    int b = o2 * 8 + i2;                          // [0,64)
    const float* g = g012 + ((size_t)a * 64 + b) * 16;
    float s = 0.f;
#pragma unroll
    for (int r3 = 0; r3 < 16; ++r3)
        s += g[r3] * f3[(r3 * 8 + o3) * 8 + i3];
    W[t] = s;
}

// ---------------------------------------------------------------------------
// Step 2: Y[b][o] = sum_i X[b][i] * W[o][i] + bias[o]
// Block tile 128x64, 8 waves, each wave computes a 16x64 strip via
// V_WMMA_F32_16X16X4_F32 with K staged through LDS in slabs of 32.
__global__ __launch_bounds__(256) void tt_gemm(const float* __restrict__ X,
                                               const float* __restrict__ W,
                                               const float* __restrict__ bias,
                                               float* __restrict__ Y) {
    __shared__ float sX[128 * LDSTR];
    __shared__ float sW[64 * LDSTR];

    const int tid  = threadIdx.x;
    const int wave = tid >> 5;
    const int lane = tid & 31;
    const int m0   = blockIdx.y * 128;   // batch-row tile
    const int n0   = blockIdx.x * 64;    // output-col tile
    const int lrow = lane & 15;
    const int koff = (lane >> 4) << 1;   // A/B lane K sub-offset per ISA layout

    v8f acc[4] = {};                     // 4 accumulators: 16x64 per wave

    for (int k0 = 0; k0 < DIN; k0 += 32) {
        // Cooperative load: X tile 128x32 (1024 float4, 4 per thread)
#pragma unroll
        for (int it = 0; it < 4; ++it) {
            int idx = tid + it * 256;
            int r = idx >> 3, c4 = (idx & 7) << 2;
            v4f v = *(const v4f*)(X + (size_t)(m0 + r) * DIN + k0 + c4);
            *(v4f*)(&sX[r * LDSTR + c4]) = v;
        }
        // W tile 64x32 (512 float4, 2 per thread)
#pragma unroll
        for (int it = 0; it < 2; ++it) {
            int idx = tid + it * 256;
            int r = idx >> 3, c4 = (idx & 7) << 2;
            v4f v = *(const v4f*)(W + (size_t)(n0 + r) * DIN + k0 + c4);
            *(v4f*)(&sW[r * LDSTR + c4]) = v;
        }
        __syncthreads();

        const float* xrow = &sX[(wave * 16 + lrow) * LDSTR + koff];
#pragma unroll
        for (int kk = 0; kk < 32; kk += 4) {
            v2f a = *(const v2f*)(xrow + kk);                 // A: M=lrow, K=koff+{0,1}
#pragma unroll
            for (int j = 0; j < 4; ++j) {
                v2f b = *(const v2f*)(&sW[(j * 16 + lrow) * LDSTR + koff + kk]); // B: N=lrow
                acc[j] = __builtin_amdgcn_wmma_f32_16x16x4_f32(
                    /*neg_a=*/false, a, /*neg_b=*/false, b,
                    /*c_mod=*/(short)0, acc[j], /*reuse_a=*/false, /*reuse_b=*/false);
            }
        }
        __syncthreads();
    }

    // Epilogue: C/D layout — vgpr v holds M = v + (lane>>4)*8, N = lane&15
    const int mrow = m0 + wave * 16 + (lane >> 4) * 8;
#pragma unroll
    for (int j = 0; j < 4; ++j) {
        int col  = n0 + j * 16 + lrow;
        float bv = bias[col];
#pragma unroll
        for (int v = 0; v < 8; ++v)
            Y[(size_t)(mrow + v) * DIN + col] = acc[j][v] + bv;
    }
}

// ---------------------------------------------------------------------------
extern "C" void kernel_launch(void* const* d_in, const int* in_sizes, int n_in,
                              void* d_out, int out_size, void* d_ws, size_t ws_size,
                              hipStream_t stream) {
    const float* x    = (const float*)d_in[0];
    const float* f0   = (const float*)d_in[1];
    const float* f1   = (const float*)d_in[2];
    const float* f2   = (const float*)d_in[3];
    const float* f3   = (const float*)d_in[4];
    const float* bias = (const float*)d_in[5];
    float*       out  = (float*)d_out;

    // Workspace layout (floats): W[16777216] | G01[65536] | G012[4194304]  (~84 MB)
    float* W    = (float*)d_ws;
    float* g01  = W + (size_t)DIN * DIN;
    float* g012 = g01 + 65536;

    build_g01 <<<256,   256, 0, stream>>>(f0, f1, g01);
    build_g012<<<16384, 256, 0, stream>>>(g01, f2, g012);
    build_w   <<<65536, 256, 0, stream>>>(g012, f3, W);

    dim3 grid(DIN / 64, 4096 / 128);     // (64, 32)
    tt_gemm<<<grid, 256, 0, stream>>>(x, W, bias, out);
}